// MGIoU2DPlus_4355096839047
// MI455X (gfx1250) — compile-verified
//
#include <hip/hip_runtime.h>
#include <stdint.h>

#define TPB 256

static constexpr float CONVEX_WEIGHT = 0.1f;

#if __has_builtin(__builtin_amdgcn_rcpf)
#define RCPF(x) __builtin_amdgcn_rcpf(x)
#else
#define RCPF(x) (1.0f / (x))
#endif

// Centered vertex: offset from centroid + atan2-order region id.
struct Pt { float dx, dy; int r; };

// Strict "angle(a) < angle(b)" in atan2's (-pi, pi] ascending order,
// without transcendentals. Region ids follow ascending atan2:
//   r0: dy<0 && dx<0   -> (-pi,   -pi/2)
//   r1: dy<0 && dx>=0  -> [-pi/2,  0)
//   r2: dy>=0 && dx>0  -> [0,      pi/2)
//   r3: dy>=0 && dx<=0 -> [pi/2,   pi]
// Within a region (span <= pi/2) the 2D cross product decides order.
__device__ __forceinline__ bool angLess(const Pt& a, const Pt& b) {
  if (a.r != b.r) return a.r < b.r;
  return (a.dx * b.dy - a.dy * b.dx) > 0.0f;
}

// Sort 4 vertices by angle around centroid; emit the 4 edge normals of the
// angularly-sorted polygon. Since sorted vertex = centroid + d, the edge is
// just d_next - d_cur, so we never need the absolute coordinates here.
__device__ __forceinline__ void candidateAxes(const float* x, const float* y,
                                              float* nx, float* ny) {
  const float cx = 0.25f * (x[0] + x[1] + x[2] + x[3]);
  const float cy = 0.25f * (y[0] + y[1] + y[2] + y[3]);
  Pt q[4];
#pragma unroll
  for (int i = 0; i < 4; ++i) {
    const float dx = x[i] - cx, dy = y[i] - cy;
    q[i].dx = dx; q[i].dy = dy;
    q[i].r = (dy < 0.0f) ? ((dx < 0.0f) ? 0 : 1) : ((dx > 0.0f) ? 2 : 3);
  }
  // Optimal 5-comparison sorting network for n=4; compile-time indices only,
  // 3 cndmask-selected values per exchange.
#define CE_(i, j) { if (angLess(q[j], q[i])) { Pt t = q[i]; q[i] = q[j]; q[j] = t; } }
  CE_(0, 1) CE_(2, 3) CE_(0, 2) CE_(1, 3) CE_(1, 2)
#undef CE_
#pragma unroll
  for (int k = 0; k < 4; ++k) {
    const int kn = (k + 1) & 3;
    nx[k] = q[kn].dy - q[k].dy;       //  edge.y
    ny[k] = -(q[kn].dx - q[k].dx);    // -edge.x
  }
}

__global__ __launch_bounds__(TPB) void mgiou2dplus_kernel(
    const float* __restrict__ pred, const float* __restrict__ targ,
    float* __restrict__ out, int B) {
  // 4 x float4 per thread: 8 pred + 8 target floats = 64 B/lane, 16 KB/block.
  // float4 type guarantees 16-byte alignment -> ds_load_b128 consumption.
  __shared__ float4 lds4[TPB * 4];

  const int tid = threadIdx.x;
  const long long base = (long long)blockIdx.x * TPB;
  long long e = base + tid;
  const bool live = (e < B);
  if (!live) e = (long long)B - 1;   // clamp: issue in-range loads, skip store

  // ---- CDNA5 async global->LDS staging (ASYNCcnt path) ----------------
  // IOFFSET is added to BOTH the global and the LDS address, so one address
  // pair covers two b128 chunks. Each lane gathers its own 64 B.
  const uint64_t gp = (uint64_t)(uintptr_t)(pred + e * 8);
  const uint64_t gt = (uint64_t)(uintptr_t)(targ + e * 8);
  const uint32_t lp = (uint32_t)(uintptr_t)(&lds4[tid * 4]);  // low 32 bits = LDS offset
  const uint32_t lt = lp + 32u;
  asm volatile("global_load_async_to_lds_b128 %0, %1, off"            :: "v"(lp), "v"(gp) : "memory");
  asm volatile("global_load_async_to_lds_b128 %0, %1, off offset:16"  :: "v"(lp), "v"(gp) : "memory");
  asm volatile("global_load_async_to_lds_b128 %0, %1, off"            :: "v"(lt), "v"(gt) : "memory");
  asm volatile("global_load_async_to_lds_b128 %0, %1, off offset:16"  :: "v"(lt), "v"(gt) : "memory");
  asm volatile("s_wait_asynccnt 0x0" ::: "memory");
  // ---------------------------------------------------------------------

  const float4 v0 = lds4[tid * 4 + 0];
  const float4 v1 = lds4[tid * 4 + 1];
  const float4 v2 = lds4[tid * 4 + 2];
  const float4 v3 = lds4[tid * 4 + 3];

  float px[4], py[4], tx[4], ty[4];
  px[0] = v0.x; py[0] = v0.y; px[1] = v0.z; py[1] = v0.w;
  px[2] = v1.x; py[2] = v1.y; px[3] = v1.z; py[3] = v1.w;
  tx[0] = v2.x; ty[0] = v2.y; tx[1] = v2.z; ty[1] = v2.w;
  tx[2] = v3.x; ty[2] = v3.y; tx[3] = v3.z; ty[3] = v3.w;

  // 8 separating axes: 4 from pred's sorted edges, 4 from target's.
  float ax[8], ay[8];
  candidateAxes(px, py, ax, ay);
  candidateAxes(tx, ty, ax + 4, ay + 4);

  // Per-axis 1-D GIoU:  inter/union - (hull-union)/hull = inter/union - 1 + union/hull
  float acc = 0.0f;
#pragma unroll
  for (int a = 0; a < 8; ++a) {
    const float X = ax[a], Y = ay[a];
    float p = fmaf(px[0], X, py[0] * Y);
    float mn1 = p, mx1 = p;
#pragma unroll
    for (int i = 1; i < 4; ++i) {
      p = fmaf(px[i], X, py[i] * Y);
      mn1 = fminf(mn1, p); mx1 = fmaxf(mx1, p);
    }
    float q = fmaf(tx[0], X, ty[0] * Y);
    float mn2 = q, mx2 = q;
#pragma unroll
    for (int i = 1; i < 4; ++i) {
      q = fmaf(tx[i], X, ty[i] * Y);
      mn2 = fminf(mn2, q); mx2 = fmaxf(mx2, q);
    }
    const float inter = fmaxf(fminf(mx1, mx2) - fmaxf(mn1, mn2), 0.0f);
    const float uni   = (mx1 - mn1) + (mx2 - mn2) - inter;
    const float hull  = fmaxf(mx1, mx2) - fminf(mn1, mn2);
    acc += fmaf(inter, RCPF(uni), uni * RCPF(hull)) - 1.0f;
  }
  float loss = (1.0f - acc * 0.125f) * 0.5f;

  // Convexity penalty on the *unsorted* pred polygon.
  float cr[4];
#pragma unroll
  for (int i = 0; i < 4; ++i) {
    const int ip = (i + 3) & 3, in_ = (i + 1) & 3;
    const float e1x = px[ip] - px[i], e1y = py[ip] - py[i];
    const float e2x = px[in_] - px[i], e2y = py[in_] - py[i];
    cr[i] = e1x * e2y - e1y * e2x;
  }
  const float sref = (cr[0] < 0.0f) ? -1.0f : 1.0f;  // sign(c0), 0 -> +1
  float pen = 0.0f;
#pragma unroll
  for (int i = 0; i < 4; ++i) pen += fmaxf(-sref * cr[i], 0.0f);
  loss += CONVEX_WEIGHT * 0.25f * pen;

  if (live) out[base + tid] = loss;
}

extern "C" void kernel_launch(void* const* d_in, const int* in_sizes, int n_in,
                              void* d_out, int out_size, void* d_ws, size_t ws_size,
                              hipStream_t stream) {
  (void)n_in; (void)d_ws; (void)ws_size; (void)out_size;
  const float* pred = (const float*)d_in[0];
  const float* targ = (const float*)d_in[1];
  float* out = (float*)d_out;
  const int B = in_sizes[0] / 8;           // [B,4,2] f32
  const int grid = (B + TPB - 1) / TPB;
  mgiou2dplus_kernel<<<grid, TPB, 0, stream>>>(pred, targ, out, B);
}